// diag_rtrl_61151744360732
// MI455X (gfx1250) — compile-verified
//
#include <hip/hip_runtime.h>
#include <cmath>

// Problem constants (match reference: B=32, D=1024, H=1024, BETA=0.9)
#define B_    32
#define D_    1024
#define H_    1024
#define BETA_ 0.9f

typedef __attribute__((ext_vector_type(2))) float v2f;
typedef __attribute__((ext_vector_type(4))) float v4f;
typedef __attribute__((ext_vector_type(8))) float v8f;

// ---------------------------------------------------------------------------
// Kernel 1: z = x @ w via V_WMMA_F32_16X16X4_F32, fused epilogue:
//   u_new = 0.9*u + z + b ; out = tanh(u_new) ; E_b_new = 0.9*E_b + 1
// One wave32 per 16x16 output tile. 128 tiles = 32 blocks * 4 waves.
// K loop: 1024 / 4 = 256 WMMA steps per wave.
//
// fp32 WMMA fragment layouts (CDNA5 ISA 7.12.2):
//   A (16x4):  lanes 0-15 hold M=lane, K={0,1} in v[0],v[1];
//              lanes 16-31 hold M=lane-16, K={2,3}.
//   B (4x16):  lanes 0-15 hold N=lane, K={0,1} in v[0],v[1];
//              lanes 16-31 hold N=lane-16, K={2,3}.
//   C/D 16x16: VGPR r -> M = r + 8*(lane>>4), N = lane&15.
// ---------------------------------------------------------------------------
__global__ __launch_bounds__(128) void rtrl_gemm_epilogue_kernel(
    const float* __restrict__ x,     // [B, D]
    const float* __restrict__ w,     // [D, H]
    const float* __restrict__ bias,  // [H]
    const float* __restrict__ u,     // [B, H]
    const float* __restrict__ E_b,   // [B, H] (leading 1 dim dropped)
    float* __restrict__ out,         // [B, H]
    float* __restrict__ u_new,       // [B, H]
    float* __restrict__ E_b_new)     // [B, H]
{
    const int lane  = threadIdx.x & 31;
    const int wave  = threadIdx.x >> 5;
    const int tile  = blockIdx.x * 4 + wave;   // 0..127
    const int mtile = tile & 1;                // 2 row tiles (B=32)
    const int ntile = tile >> 1;               // 64 col tiles (H=1024)
    const int m0    = mtile * 16;
    const int n0    = ntile * 16;

    const int row   = lane & 15;   // M (for A) / N (for B) within tile
    const int khalf = lane >> 4;   // selects K pair {0,1} vs {2,3}

    v8f acc = {};
    const float* xrow = x + (size_t)(m0 + row) * D_;

    for (int k = 0; k < D_; k += 4) {
        // A fragment: two consecutive K values -> one 8-byte load
        v2f afrag = *(const v2f*)(xrow + k + 2 * khalf);
        // B fragment: w is row-major [D,H]; rows k+2*khalf and k+2*khalf+1
        v2f bfrag;
        bfrag.x = w[(size_t)(k + 2 * khalf)     * H_ + n0 + row];
        bfrag.y = w[(size_t)(k + 2 * khalf + 1) * H_ + n0 + row];
        // D = A*B + C  (8 args: neg_a, A, neg_b, B, c_mod, C, reuse_a, reuse_b)
        acc = __builtin_amdgcn_wmma_f32_16x16x4_f32(
            false, afrag, false, bfrag, (short)0, acc, false, false);
    }

    // Epilogue: each VGPR r of the accumulator is element (m0+r+8*khalf, n0+col)
    const int col = lane & 15;
    const int hj  = n0 + col;
    const float bj = bias[hj];
#pragma unroll
    for (int r = 0; r < 8; ++r) {
        const int    mi  = m0 + r + 8 * khalf;
        const size_t idx = (size_t)mi * H_ + hj;
        const float  un  = BETA_ * u[idx] + acc[r] + bj;
        u_new[idx]   = un;
        out[idx]     = tanhf(un);
        E_b_new[idx] = BETA_ * E_b[idx] + 1.0f;
    }
}

// ---------------------------------------------------------------------------
// Kernel 2 (bandwidth-dominant, ~268 MB of HBM traffic -> ~11.5 us @ 23.3 TB/s):
//   E_w_new[i,d,h] = 0.9 * E_w[i,d,h] + x[i,d]
// One block per (i,d) row; 256 threads x float4 = 1024 floats (= H).
// Non-temporal loads/stores: pure stream, keep it out of L2.
// ---------------------------------------------------------------------------
__global__ __launch_bounds__(256) void rtrl_ew_update_kernel(
    const float* __restrict__ E_w,   // [B*D, H]
    const float* __restrict__ x,     // [B*D]
    float* __restrict__ E_w_new)     // [B*D, H]
{
    const int    rowid = blockIdx.x;              // i*D + d
    const float  xv    = x[rowid];                // uniform -> scalar load
    const size_t base  = (size_t)rowid * H_ + (size_t)threadIdx.x * 4;

    const v4f e = __builtin_nontemporal_load((const v4f*)(E_w + base));
    v4f o;
    o.x = BETA_ * e.x + xv;
    o.y = BETA_ * e.y + xv;
    o.z = BETA_ * e.z + xv;
    o.w = BETA_ * e.w + xv;
    __builtin_nontemporal_store(o, (v4f*)(E_w_new + base));
}

// ---------------------------------------------------------------------------
// Launch. Inputs (setup_inputs order): x, w, b, u, E_w, E_b.
// Output layout (flat, return order): out[32K] | u_new[32K] | E_w_new[32M] | E_b_new[32K]
// ---------------------------------------------------------------------------
extern "C" void kernel_launch(void* const* d_in, const int* in_sizes, int n_in,
                              void* d_out, int out_size, void* d_ws, size_t ws_size,
                              hipStream_t stream) {
    (void)in_sizes; (void)n_in; (void)out_size; (void)d_ws; (void)ws_size;

    const float* x   = (const float*)d_in[0];
    const float* w   = (const float*)d_in[1];
    const float* b   = (const float*)d_in[2];
    const float* u   = (const float*)d_in[3];
    const float* E_w = (const float*)d_in[4];
    const float* E_b = (const float*)d_in[5];

    float* out     = (float*)d_out;
    float* u_new   = out + (size_t)B_ * H_;
    float* E_w_new = u_new + (size_t)B_ * H_;
    float* E_b_new = E_w_new + (size_t)B_ * D_ * H_;

    // Small fused GEMM (WMMA fp32) + pointwise epilogue: 128 waves total.
    rtrl_gemm_epilogue_kernel<<<32, 128, 0, stream>>>(
        x, w, b, u, E_b, out, u_new, E_b_new);

    // Dominant streaming update of the eligibility trace.
    rtrl_ew_update_kernel<<<B_ * D_, 256, 0, stream>>>(E_w, x, E_w_new);
}